// SpectralConv2d_6519760355683
// MI455X (gfx1250) — compile-verified
//
#include <hip/hip_runtime.h>
#include <math.h>

// ---------------------------------------------------------------------------
// Problem constants (from reference)
// ---------------------------------------------------------------------------
constexpr int Bn = 8, CIN = 64, COUT = 64, Hn = 256, Wn = 256, M1 = 20, M2 = 20;
constexpr int NIMG = Bn * CIN;          // 512 input images (== Bn*COUT output images)
constexpr int NROW = 64;                // padded corner rows: [0,20)=corner0, [32,52)=corner1

// ---------------------------------------------------------------------------
// Workspace layout (floats), each region tail-padded by 64 floats
// ---------------------------------------------------------------------------
constexpr size_t PAD     = 64;
constexpr size_t OFF_TW  = 0;                                   // Tw  [256][32]  forward W cas (cols>=20 zero)
constexpr size_t OFF_TWS = OFF_TW  + 256 * 32 + PAD;            // TwS [20][256]  inverse W cas * 2^-24
constexpr size_t OFF_THA = OFF_TWS + 20 * 256 + PAD;            // ThA [64][256]  forward H cas * 1/256 (pad rows zero)
constexpr size_t OFF_THB = OFF_THA + 64 * 256 + PAD;            // ThB [256][64]  inverse H cas (pad cols zero)
constexpr size_t OFF_XD  = OFF_THB + 256 * 64 + PAD;            // XD  [512][64][20]
constexpr size_t OFF_OM  = OFF_XD  + (size_t)NIMG * NROW * 20 + PAD; // OM [512][64][20]
constexpr size_t OFF_T1  = OFF_OM  + (size_t)NIMG * NROW * 20 + PAD; // T1 [512][256][20]
constexpr size_t OFF_Y1  = OFF_T1  + (size_t)NIMG * 256 * 20 + PAD;  // Y1 [512][256][20]
constexpr size_t WS_FLOATS = OFF_Y1 + (size_t)NIMG * 256 * 20 + PAD;

// ---------------------------------------------------------------------------
// Async global->LDS path (CDNA5 GLOBAL_LOAD_ASYNC_TO_LDS_B128 + ASYNCcnt),
// guarded so absence of the builtins falls back to the VALU/DS staging path.
// ---------------------------------------------------------------------------
#if defined(__HIP_DEVICE_COMPILE__) &&                                        \
    __has_builtin(__builtin_amdgcn_global_load_async_to_lds_b128) &&          \
    __has_builtin(__builtin_amdgcn_s_wait_asynccnt)
#define FNO_ASYNC_LDS 1
#else
#define FNO_ASYNC_LDS 0
#endif

typedef __attribute__((ext_vector_type(4))) int v4i;
typedef __attribute__((address_space(1))) v4i* as1_v4i_p;  // global ("__device__")
typedef __attribute__((address_space(3))) v4i* as3_v4i_p;  // LDS ("__shared__")

#if FNO_ASYNC_LDS
// Compile-time-unrolled async copy: N issues of B128, immediate offset I*1024
// (per ISA, INST_OFFSET is added to BOTH the LDS and global addresses).
template <int I, int N>
struct AsyncCopy {
  static __device__ __forceinline__ void run(as1_v4i_p g, as3_v4i_p l) {
    __builtin_amdgcn_global_load_async_to_lds_b128(g, l, I * 1024, 0);
    AsyncCopy<I + 1, N>::run(g, l);
  }
};
template <int N>
struct AsyncCopy<N, N> {
  static __device__ __forceinline__ void run(as1_v4i_p, as3_v4i_p) {}
};
#endif

// ---------------------------------------------------------------------------
// CDNA5 f32 WMMA: D(16x16,f32) = A(16x4,f32) * B(4x16,f32) + C
//   A frag: lanes 0-15 -> M=lane, vgpr j -> K=j;  lanes 16-31 -> K=2+j
//   B frag: vgpr j, lanes 0-15 -> K=j, N=lane;    lanes 16-31 -> K=2+j
//   C frag: vgpr v, lanes 0-15 -> M=v, N=lane;    lanes 16-31 -> M=v+8
// ---------------------------------------------------------------------------
typedef __attribute__((ext_vector_type(2))) float v2f;
typedef __attribute__((ext_vector_type(8))) float v8f;

static __device__ __forceinline__ v8f wmma4(v2f a, v2f b, v8f c) {
  return __builtin_amdgcn_wmma_f32_16x16x4_f32(false, a, false, b, (short)0, c,
                                               false, false);
}

// load A fragment from row-major src (row stride ld): rows m0.., K offset k0
static __device__ __forceinline__ v2f ldA(const float* __restrict__ s, int ld,
                                          int m0, int k0, int lo, int hi) {
  v2f a;
  const float* p = s + (size_t)(m0 + lo) * ld + k0 + 2 * hi;
  a[0] = p[0];
  a[1] = p[1];
  return a;
}
// load B fragment from row-major src (row stride ld): K rows k0.., cols n0..
static __device__ __forceinline__ v2f ldB(const float* __restrict__ s, int ld,
                                          int k0, int n0, int lo, int hi) {
  v2f b;
  const float* p = s + (size_t)(k0 + 2 * hi) * ld + n0 + lo;
  b[0] = p[0];
  b[1] = p[ld];
  return b;
}

// ---------------------------------------------------------------------------
// Kernel 0: build cas tables + zero OM (deterministic per call)
// ---------------------------------------------------------------------------
__global__ void fno_init(float* __restrict__ ws) {
  const float STEP = 6.283185307179586f / 256.0f;
  int idx = blockIdx.x * blockDim.x + threadIdx.x;
  int stride = gridDim.x * blockDim.x;

  for (int t = idx; t < 256 * 32; t += stride) {            // Tw [n2][k2]
    int n = t >> 5, k = t & 31;
    float v = 0.0f;
    if (k < M2) { float s, c; __sincosf(STEP * (float)((n * k) & 255), &s, &c); v = c + s; }
    ws[OFF_TW + t] = v;
  }
  for (int t = idx; t < 20 * 256; t += stride) {            // TwS [k2][n2] * 2^-24
    int k = t >> 8, n = t & 255;
    float s, c; __sincosf(STEP * (float)((n * k) & 255), &s, &c);
    ws[OFF_TWS + t] = (c + s) * (1.0f / 16777216.0f);
  }
  for (int t = idx; t < 64 * 256; t += stride) {            // ThA [r][n1] * 1/256
    int r = t >> 8, n = t & 255;
    int k1 = (r < 20) ? r : ((r >= 32 && r < 52) ? (204 + r) : -1);
    float v = 0.0f;
    if (k1 >= 0) { float s, c; __sincosf(STEP * (float)((n * k1) & 255), &s, &c); v = (c + s) * (1.0f / 256.0f); }
    ws[OFF_THA + t] = v;
  }
  for (int t = idx; t < 256 * 64; t += stride) {            // ThB [n1][r]
    int n = t >> 6, r = t & 63;
    int k1 = (r < 20) ? r : ((r >= 32 && r < 52) ? (204 + r) : -1);
    float v = 0.0f;
    if (k1 >= 0) { float s, c; __sincosf(STEP * (float)((n * k1) & 255), &s, &c); v = c + s; }
    ws[OFF_THB + t] = v;
  }
  for (int t = idx; t < NIMG * NROW * 20; t += stride) ws[OFF_OM + t] = 0.0f;
}

// ---------------------------------------------------------------------------
// Stage 1: T1[img][n1][k2] = sum_n2 x[img][n1][n2] * Tw[n2][k2]
// grid (8, 512), block 64 (2 waves). 32 x-rows staged in LDS via the CDNA5
// async global->LDS DMA path (ASYNCcnt) when available.
// ---------------------------------------------------------------------------
__global__ __launch_bounds__(64) void fno_stage1(const float* __restrict__ x,
                                                 const float* __restrict__ ws) {
  __shared__ __align__(16) float xs[32][256];
  const float* Tw = ws + OFF_TW;
  float* T1 = (float*)(ws + OFF_T1);

  int img = blockIdx.y, rb = blockIdx.x;
  const float* xim = x + (size_t)img * (Hn * Wn) + (size_t)rb * 32 * 256;
  int tid = threadIdx.x;

#if FNO_ASYNC_LDS
  {
    // per-thread: 32 async B128 copies, stride 1024B (block-wide coalesced)
    as1_v4i_p g = (as1_v4i_p)(xim + (size_t)tid * 4);
    as3_v4i_p l = (as3_v4i_p)(&xs[0][0] + (size_t)tid * 4);
    AsyncCopy<0, 32>::run(g, l);
    __builtin_amdgcn_s_wait_asynccnt(0);
  }
  __syncthreads();
#else
  {
    const float4* src4 = (const float4*)xim;
    float4* dst4 = (float4*)&xs[0][0];
    for (int i = tid; i < 32 * 64; i += 64) dst4[i] = src4[i];
  }
  __syncthreads();
#endif

  int wave = tid >> 5, lane = tid & 31, lo = lane & 15, hi = lane >> 4;
  const float* A = &xs[wave * 16][0];
  v8f acc0 = {}, acc1 = {};
  for (int k0 = 0; k0 < 256; k0 += 4) {
    v2f a = ldA(A, 256, 0, k0, lo, hi);
    v2f b0 = ldB(Tw, 32, k0, 0, lo, hi);
    v2f b1 = ldB(Tw, 32, k0, 16, lo, hi);
    acc0 = wmma4(a, b0, acc0);
    acc1 = wmma4(a, b1, acc1);
  }
  float* o = T1 + (size_t)img * (256 * 20) + (size_t)(rb * 32 + wave * 16) * 20;
#pragma unroll
  for (int v = 0; v < 8; v++) {
    int m = v + 8 * hi;
    o[m * 20 + lo] = acc0[v];
    if (lo < 4) o[m * 20 + 16 + lo] = acc1[v];
  }
}

// ---------------------------------------------------------------------------
// Stage 2: XD[img][r][k2] = sum_n1 ThA[r][n1] * T1[img][n1][k2]
// grid (4, 512), block 32 (one wave per 16-row m-tile of r)
// ---------------------------------------------------------------------------
__global__ __launch_bounds__(32) void fno_stage2(float* __restrict__ ws) {
  const float* ThA = ws + OFF_THA;
  const float* T1 = ws + OFF_T1;
  float* XD = ws + OFF_XD;

  int img = blockIdx.y, mt = blockIdx.x;
  int lane = threadIdx.x, lo = lane & 15, hi = lane >> 4;
  const float* Bm = T1 + (size_t)img * (256 * 20);
  v8f acc0 = {}, acc1 = {};
  for (int k0 = 0; k0 < 256; k0 += 4) {
    v2f a = ldA(ThA, 256, mt * 16, k0, lo, hi);
    v2f b0 = ldB(Bm, 20, k0, 0, lo, hi);
    v2f b1 = ldB(Bm, 20, k0, 16, lo, hi);   // cols 20..31 garbage -> discarded
    acc0 = wmma4(a, b0, acc0);
    acc1 = wmma4(a, b1, acc1);
  }
  float* o = XD + (size_t)img * (NROW * 20) + (size_t)(mt * 16) * 20;
#pragma unroll
  for (int v = 0; v < 8; v++) {
    int m = v + 8 * hi;
    o[m * 20 + lo] = acc0[v];
    if (lo < 4) o[m * 20 + 16 + lo] = acc1[v];
  }
}

// ---------------------------------------------------------------------------
// Mode mixing:  OM[b*64+o][r][k2] = sum_i XD*we + XDn*wo  per mode (corner,a,b)
// we = 0.5(w + wn), wo = 0.5(w - wn), wn = w at ((-a)%20, (-b)%20)
// grid 800 = 2 corners * 400 modes, block 256
// ---------------------------------------------------------------------------
__global__ __launch_bounds__(256) void fno_modes(const float* __restrict__ w1,
                                                 const float* __restrict__ w2,
                                                 float* __restrict__ ws) {
  __shared__ float Xs[8][64];
  __shared__ float Xn[8][64];
  const float* XD = ws + OFF_XD;
  float* OM = ws + OFF_OM;

  int blk = blockIdx.x;
  int c = blk / 400, mode = blk - c * 400;
  int a = mode / 20, b = mode % 20;
  int an = (20 - a) % 20, bn = (20 - b) % 20;
  int r = c * 32 + a, rn = c * 32 + an;
  int tid = threadIdx.x;

  for (int t = tid; t < 512; t += 256) {
    int bm = t >> 6, i = t & 63;
    size_t im = (size_t)(bm * CIN + i) * (NROW * 20);
    Xs[bm][i] = XD[im + r * 20 + b];
    Xn[bm][i] = XD[im + rn * 20 + bn];
  }
  __syncthreads();

  const float* w = (c == 0) ? w1 : w2;
  for (int t = tid; t < 512; t += 256) {
    int bm = t >> 6, o = t & 63;
    float acc = 0.0f;
#pragma unroll 4
    for (int i = 0; i < 64; i++) {
      float wv = w[((size_t)(i * 64 + o) * 20 + a) * 20 + b];
      float wn = w[((size_t)(i * 64 + o) * 20 + an) * 20 + bn];
      acc += Xs[bm][i] * (0.5f * (wv + wn)) + Xn[bm][i] * (0.5f * (wv - wn));
    }
    OM[(size_t)(bm * COUT + o) * (NROW * 20) + r * 20 + b] = acc;
  }
}

// ---------------------------------------------------------------------------
// Stage 3: Y1[img][n1][k2] = sum_r ThB[n1][r] * OM[img][r][k2]
// grid (16, 512), block 32
// ---------------------------------------------------------------------------
__global__ __launch_bounds__(32) void fno_stage3(float* __restrict__ ws) {
  const float* ThB = ws + OFF_THB;
  const float* OM = ws + OFF_OM;
  float* Y1 = ws + OFF_Y1;

  int img = blockIdx.y, mt = blockIdx.x;
  int lane = threadIdx.x, lo = lane & 15, hi = lane >> 4;
  const float* Bm = OM + (size_t)img * (NROW * 20);
  v8f acc0 = {}, acc1 = {};
  for (int k0 = 0; k0 < 64; k0 += 4) {
    v2f a = ldA(ThB, 64, mt * 16, k0, lo, hi);
    v2f b0 = ldB(Bm, 20, k0, 0, lo, hi);
    v2f b1 = ldB(Bm, 20, k0, 16, lo, hi);
    acc0 = wmma4(a, b0, acc0);
    acc1 = wmma4(a, b1, acc1);
  }
  float* o = Y1 + (size_t)img * (256 * 20) + (size_t)(mt * 16) * 20;
#pragma unroll
  for (int v = 0; v < 8; v++) {
    int m = v + 8 * hi;
    o[m * 20 + lo] = acc0[v];
    if (lo < 4) o[m * 20 + 16 + lo] = acc1[v];
  }
}

// ---------------------------------------------------------------------------
// Stage 4 (fused): xr = Y1 @ TwS (two mirror 16-row bands in LDS), then
//   out[n1][n2] = atan2(xr[255-n1][255-n2], xr[n1][n2])
// grid (8, 512), block 128 (4 waves). Band pair p and 15-p.
// ---------------------------------------------------------------------------
__global__ __launch_bounds__(128) void fno_stage4(const float* __restrict__ ws,
                                                  float* __restrict__ out) {
  __shared__ float xr[2][16][260];
  const float* TwS = ws + OFF_TWS;
  const float* Y1 = ws + OFF_Y1;

  int img = blockIdx.y, p = blockIdx.x;
  int tid = threadIdx.x, wave = tid >> 5, lane = tid & 31;
  int lo = lane & 15, hi = lane >> 4;
  const float* Yim = Y1 + (size_t)img * (256 * 20);

  for (int tI = wave; tI < 32; tI += 4) {   // 2 halves x 16 n-tiles
    int h = tI >> 4, nt = tI & 15;
    int mrow = (h == 0 ? p : 15 - p) * 16;
    v8f acc = {};
    for (int k0 = 0; k0 < 20; k0 += 4) {
      v2f a = ldA(Yim, 20, mrow, k0, lo, hi);
      v2f b = ldB(TwS, 256, k0, nt * 16, lo, hi);
      acc = wmma4(a, b, acc);
    }
#pragma unroll
    for (int v = 0; v < 8; v++) xr[h][v + 8 * hi][nt * 16 + lo] = acc[v];
  }
  __syncthreads();

  float* oim = out + (size_t)img * (Hn * Wn);
  for (int t = tid; t < 32 * 256; t += 128) {
    int half = t >> 12, rr = (t >> 8) & 15, n2 = t & 255;
    int n1 = (half == 0 ? p : 15 - p) * 16 + rr;
    float cosv = xr[half][rr][n2];
    float sinv = xr[half ^ 1][15 - rr][255 - n2];
    oim[n1 * 256 + n2] = atan2f(sinv, cosv);
  }
}

// ---------------------------------------------------------------------------
// Host launcher
// ---------------------------------------------------------------------------
extern "C" void kernel_launch(void* const* d_in, const int* in_sizes, int n_in,
                              void* d_out, int out_size, void* d_ws, size_t ws_size,
                              hipStream_t stream) {
  (void)in_sizes; (void)n_in; (void)out_size;
  if (ws_size < WS_FLOATS * sizeof(float)) return;

  const float* x = (const float*)d_in[0];
  const float* w1 = (const float*)d_in[1];
  const float* w2 = (const float*)d_in[2];
  float* ws = (float*)d_ws;
  float* out = (float*)d_out;

  fno_init<<<256, 256, 0, stream>>>(ws);
  fno_stage1<<<dim3(8, NIMG), 64, 0, stream>>>(x, ws);
  fno_stage2<<<dim3(4, NIMG), 32, 0, stream>>>(ws);
  fno_modes<<<800, 256, 0, stream>>>(w1, w2, ws);
  fno_stage3<<<dim3(16, NIMG), 32, 0, stream>>>(ws);
  fno_stage4<<<dim3(8, NIMG), 128, 0, stream>>>(ws, out);
}